// TGCN_83202106458535
// MI455X (gfx1250) — compile-verified
//
#include <hip/hip_runtime.h>
#include <hip/hip_bf16.h>

#define N_NODES 100000
#define N_EDGES 220000
#define HID     128
#define NG      4096
#define PE_DIM  5

typedef __attribute__((ext_vector_type(2))) float v2f;
typedef __attribute__((ext_vector_type(8))) float v8f;

// ---------------------------------------------------------------------------
// x = embed[tok] + pe @ Wt + bt     (one block per node, 128 threads)
// ---------------------------------------------------------------------------
__global__ __launch_bounds__(HID) void k_compute_x(
    const int* __restrict__ tok, const float* __restrict__ pe,
    const float* __restrict__ emb, const float* __restrict__ Wt,
    const float* __restrict__ bt, float* __restrict__ x) {
  const int n = blockIdx.x;
  const int h = threadIdx.x;
  const float* per = pe + (size_t)n * PE_DIM;
  float v = emb[(size_t)tok[n] * HID + h] + bt[h];
  v += per[0] * Wt[0 * HID + h];
  v += per[1] * Wt[1 * HID + h];
  v += per[2] * Wt[2 * HID + h];
  v += per[3] * Wt[3 * HID + h];
  v += per[4] * Wt[4 * HID + h];
  x[(size_t)n * HID + h] = v;
}

// ---------------------------------------------------------------------------
// H[N,128] = X[N,128] @ W[128,128]  via V_WMMA_F32_16X16X4_F32.
// One wave per 16-row tile; 8 column tiles (full 128 cols) held in registers.
// ISA 7.12.2 32-bit layouts:
//   A 16x4 : lanes 0-15 rows, VGPR0=K0,VGPR1=K1; lanes 16-31: K2,K3
//   B 4x16 : lanes 0-15 cols, VGPR0=K0,VGPR1=K1; lanes 16-31: K2,K3
//   C 16x16: lanes 0-15 cols, VGPR r -> M=r (lo half) / M=r+8 (hi half)
// ---------------------------------------------------------------------------
__global__ __launch_bounds__(32) void k_gemm_xw(
    const float* __restrict__ X, const float* __restrict__ W,
    float* __restrict__ Hout) {
  const int row0 = blockIdx.x * 16;
  const int lane = threadIdx.x;
  const int half = lane >> 4;       // 0: K{0,1}, 1: K{2,3}
  const int l16  = lane & 15;

  v8f acc[8];
#pragma unroll
  for (int t = 0; t < 8; ++t) acc[t] = (v8f){};

  const float* xrow = X + (size_t)(row0 + l16) * HID;
  for (int k = 0; k < HID; k += 4) {
    v2f a;
    a.x = xrow[k + half * 2 + 0];
    a.y = xrow[k + half * 2 + 1];
#pragma unroll
    for (int t = 0; t < 8; ++t) {
      const int col = t * 16 + l16;
      v2f b;
      b.x = W[(size_t)(k + half * 2 + 0) * HID + col];
      b.y = W[(size_t)(k + half * 2 + 1) * HID + col];
      acc[t] = __builtin_amdgcn_wmma_f32_16x16x4_f32(
          false, a, false, b, (short)0, acc[t], false, false);
    }
  }

#pragma unroll
  for (int t = 0; t < 8; ++t) {
    const int col = t * 16 + l16;
#pragma unroll
    for (int r = 0; r < 8; ++r) {
      const int row = row0 + r + half * 8;
      Hout[(size_t)row * HID + col] = acc[t][r];
    }
  }
}

// ---------------------------------------------------------------------------
// small elementwise / scatter kernels
// ---------------------------------------------------------------------------
__global__ void k_fill(float* __restrict__ p, float val, int n) {
  int i = blockIdx.x * blockDim.x + threadIdx.x;
  if (i < n) p[i] = val;
}

__global__ void k_deg_count(const int* __restrict__ dst, float* __restrict__ deg,
                            int e) {
  int i = blockIdx.x * blockDim.x + threadIdx.x;
  if (i < e) atomicAdd(&deg[dst[i]], 1.0f);
}

__global__ void k_rsqrt(float* __restrict__ p, int n) {
  int i = blockIdx.x * blockDim.x + threadIdx.x;
  if (i < n) p[i] = rsqrtf(p[i]);
}

// agg = h * dinv^2  (self-loop term, also re-initializes agg each call)
__global__ __launch_bounds__(HID) void k_agg_init(
    const float* __restrict__ h, const float* __restrict__ dinv,
    float* __restrict__ agg) {
  const int n = blockIdx.x;
  const int c = threadIdx.x;
  const float di = dinv[n];
  agg[(size_t)n * HID + c] = h[(size_t)n * HID + c] * di * di;
}

// agg[dst] += h[src] * dinv[src]*dinv[dst]   (one block per edge)
__global__ __launch_bounds__(HID) void k_edge_scatter_h(
    const int* __restrict__ src, const int* __restrict__ dst,
    const float* __restrict__ h, const float* __restrict__ dinv,
    float* __restrict__ agg) {
  const int e = blockIdx.x;
  const int c = threadIdx.x;
  const int s = src[e], d = dst[e];
  const float coef = dinv[s] * dinv[d];
  atomicAdd(&agg[(size_t)d * HID + c], h[(size_t)s * HID + c] * coef);
}

__global__ __launch_bounds__(HID) void k_relu_bias(
    const float* __restrict__ agg, const float* __restrict__ b1,
    float* __restrict__ hr) {
  const int n = blockIdx.x;
  const int c = threadIdx.x;
  float v = agg[(size_t)n * HID + c] + b1[c];
  hr[(size_t)n * HID + c] = v > 0.0f ? v : 0.0f;
}

// s1[n] = dot(hr[n,:], w2)   one wave32 per node, 8 waves per block
__global__ __launch_bounds__(256) void k_dot_w2(
    const float* __restrict__ hr, const float* __restrict__ w2,
    float* __restrict__ s1, int n_nodes) {
  const int wid  = threadIdx.x >> 5;
  const int lane = threadIdx.x & 31;
  const int n = blockIdx.x * 8 + wid;
  if (n >= n_nodes) return;
  const float* hp = hr + (size_t)n * HID;
  float s = 0.0f;
#pragma unroll
  for (int j = 0; j < 4; ++j) s += hp[lane + j * 32] * w2[lane + j * 32];
#pragma unroll
  for (int off = 16; off > 0; off >>= 1) s += __shfl_down(s, off, 32);
  if (lane == 0) s1[n] = s;
}

__global__ void k_s2_init(const float* __restrict__ s1,
                          const float* __restrict__ dinv,
                          float* __restrict__ s2, int n) {
  int i = blockIdx.x * blockDim.x + threadIdx.x;
  if (i < n) {
    float di = dinv[i];
    s2[i] = s1[i] * di * di;
  }
}

__global__ void k_edge_scatter_s(const int* __restrict__ src,
                                 const int* __restrict__ dst,
                                 const float* __restrict__ s1,
                                 const float* __restrict__ dinv,
                                 float* __restrict__ s2, int e) {
  int i = blockIdx.x * blockDim.x + threadIdx.x;
  if (i < e) {
    const int s = src[i], d = dst[i];
    atomicAdd(&s2[d], s1[s] * dinv[s] * dinv[d]);
  }
}

__global__ void k_z_acc(const float* __restrict__ s2,
                        const float* __restrict__ b2, float* __restrict__ z,
                        int n) {
  int i = blockIdx.x * blockDim.x + threadIdx.x;
  if (i < n) z[i] += s2[i] + b2[0];
}

__global__ void k_batch_scatter(const int* __restrict__ batch,
                                const float* __restrict__ z,
                                float* __restrict__ y, int n) {
  int i = blockIdx.x * blockDim.x + threadIdx.x;
  if (i < n) atomicAdd(&y[batch[i]], z[i]);
}

// ---------------------------------------------------------------------------
extern "C" void kernel_launch(void* const* d_in, const int* in_sizes, int n_in,
                              void* d_out, int out_size, void* d_ws,
                              size_t ws_size, hipStream_t stream) {
  (void)in_sizes; (void)n_in; (void)out_size; (void)ws_size;

  const int*   tok   = (const int*)d_in[0];
  const float* pe    = (const float*)d_in[1];
  const int*   ei    = (const int*)d_in[2];
  const int*   vei   = (const int*)d_in[3];
  const int*   batch = (const int*)d_in[4];
  const float* emb   = (const float*)d_in[5];
  const float* Wt    = (const float*)d_in[6];
  const float* bt    = (const float*)d_in[7];
  const float* W1    = (const float*)d_in[8];
  const float* b1    = (const float*)d_in[9];
  const float* W2    = (const float*)d_in[10];
  const float* b2    = (const float*)d_in[11];
  const float* Wv1   = (const float*)d_in[12];
  const float* bv1   = (const float*)d_in[13];
  const float* Wv2   = (const float*)d_in[14];
  const float* bv2   = (const float*)d_in[15];

  float* y = (float*)d_out;            // [G]
  float* x = (float*)d_out + NG;       // [N, H] (second tuple output)

  float* ws     = (float*)d_ws;
  float* hbuf   = ws;                                   // [N,H]
  float* aggbuf = hbuf + (size_t)N_NODES * HID;         // [N,H]
  float* dinv   = aggbuf + (size_t)N_NODES * HID;       // [N]
  float* s1     = dinv + N_NODES;                       // [N]
  float* s2     = s1 + N_NODES;                         // [N]
  float* z      = s2 + N_NODES;                         // [N]

  const int TB = 256;
  const int gN = (N_NODES + TB - 1) / TB;
  const int gE = (N_EDGES + TB - 1) / TB;

  // node features (also the second output)
  k_compute_x<<<N_NODES, HID, 0, stream>>>(tok, pe, emb, Wt, bt, x);
  // z accumulator across the two branches
  k_fill<<<gN, TB, 0, stream>>>(z, 0.0f, N_NODES);

  struct Branch {
    const int* e; const float* W1; const float* b1;
    const float* W2; const float* b2;
  };
  const Branch br[2] = {{ei,  W1,  b1,  W2,  b2},
                        {vei, Wv1, bv1, Wv2, bv2}};

  for (int b = 0; b < 2; ++b) {
    const int* src = br[b].e;
    const int* dst = br[b].e + N_EDGES;

    // dinv = rsqrt(1 + indegree)
    k_fill<<<gN, TB, 0, stream>>>(dinv, 1.0f, N_NODES);
    k_deg_count<<<gE, TB, 0, stream>>>(dst, dinv, N_EDGES);
    k_rsqrt<<<gN, TB, 0, stream>>>(dinv, N_NODES);

    // layer 1: h = x @ W1  (WMMA), then normalized aggregation + relu + bias
    k_gemm_xw<<<N_NODES / 16, 32, 0, stream>>>(x, br[b].W1, hbuf);
    k_agg_init<<<N_NODES, HID, 0, stream>>>(hbuf, dinv, aggbuf);
    k_edge_scatter_h<<<N_EDGES, HID, 0, stream>>>(src, dst, hbuf, dinv, aggbuf);
    k_relu_bias<<<N_NODES, HID, 0, stream>>>(aggbuf, br[b].b1, hbuf);

    // layer 2: scalar head + aggregation
    k_dot_w2<<<(N_NODES + 7) / 8, 256, 0, stream>>>(hbuf, br[b].W2, s1, N_NODES);
    k_s2_init<<<gN, TB, 0, stream>>>(s1, dinv, s2, N_NODES);
    k_edge_scatter_s<<<gE, TB, 0, stream>>>(src, dst, s1, dinv, s2, N_EDGES);
    k_z_acc<<<gN, TB, 0, stream>>>(s2, br[b].b2, z, N_NODES);
  }

  // graph-level pooling
  k_fill<<<(NG + TB - 1) / TB, TB, 0, stream>>>(y, 0.0f, NG);
  k_batch_scatter<<<gN, TB, 0, stream>>>(batch, z, y, N_NODES);
}